// Original_28613072126488
// MI455X (gfx1250) — compile-verified
//
#include <hip/hip_runtime.h>
#include <hip/hip_bf16.h>
#include <math.h>

typedef __attribute__((ext_vector_type(2))) float v2f;
typedef __attribute__((ext_vector_type(8))) float v8f;

#define DM       64
#define QK_COLS  256   // per node: [h0: q(64) k(64)][h1: q(64) k(64)]
#define DEG      8

// ---------------------------------------------------------------------------
// Kernel 1: QK = x @ Wqk^T + b  via V_WMMA_F32_16X16X4_F32
// Block = 512 threads (16 waves). Block -> one 16-row M tile; wave w -> N tile w.
// Output col c in [0,256): h = c>>7, W row = h*192 + (c&127)  (q at +0, k at +64).
// ---------------------------------------------------------------------------
__global__ __launch_bounds__(512)
void qk_gemm_wmma(const float* __restrict__ x,     // (L, 64)
                  const float* __restrict__ W,     // (384, 64)
                  const float* __restrict__ bias,  // (384,)
                  float* __restrict__ qk,          // (L, 256) workspace
                  int L)
{
    __shared__ float xs[16][DM + 1];               // +1 pad: conflict-free column reads

    const int row0 = blockIdx.x * 16;

    // Cooperative stage of the 16x64 x tile (1024 floats / 512 threads).
    for (int i = threadIdx.x; i < 16 * DM; i += 512) {
        int r = i >> 6, c = i & 63;
        int gr = row0 + r;
        xs[r][c] = x[(size_t)(gr < L ? gr : L - 1) * DM + c];
    }
    __syncthreads();

    const int wave = threadIdx.x >> 5;             // 0..15  -> N tile
    const int lane = threadIdx.x & 31;
    const int n    = lane & 15;                    // A-row / B-col / D-col index
    const int g    = lane >> 4;                    // K half-select
    const int col  = wave * 16 + n;                // 0..255
    const int h    = col >> 7;
    const int wrow = h * 192 + (col & 127);        // row of W_qkv supplying this col
    const float* __restrict__ wr = W + (size_t)wrow * DM;

    v8f acc = {};
    #pragma unroll
    for (int kb = 0; kb < DM; kb += 4) {
        // A 16x4 f32: lane l holds M=l%16, K = kb + 2*(l/16) + j   (ISA 7.12.2)
        v2f a, b;
        a[0] = xs[n][kb + 2 * g + 0];
        a[1] = xs[n][kb + 2 * g + 1];
        // B 4x16 f32: lane l holds N=l%16, K = kb + 2*(l/16) + j ; B[K][N] = W[wrow][K]
        b[0] = wr[kb + 2 * g + 0];
        b[1] = wr[kb + 2 * g + 1];
        acc = __builtin_amdgcn_wmma_f32_16x16x4_f32(
                  /*neg_a=*/false, a, /*neg_b=*/false, b,
                  /*c_mod=*/(short)0, acc, /*reuse_a=*/false, /*reuse_b=*/false);
    }

    const float bv = bias[wrow];
    // D 16x16 f32: VGPR r, lane l -> M = r + 8*(l/16), N = l%16
    #pragma unroll
    for (int r = 0; r < 8; ++r) {
        int m  = r + 8 * g;
        int gr = row0 + m;
        if (gr < L) qk[(size_t)gr * QK_COLS + col] = acc[r] + bv;
    }
}

// ---------------------------------------------------------------------------
// Kernel 2: per-edge scores + per-node softmax over 8 edges + head mean.
// 16 lanes per node: lane q16 -> edge j = q16&7, head h = q16>>3.
// ---------------------------------------------------------------------------
__global__ __launch_bounds__(256)
void edge_attn(const float* __restrict__ qk,      // (L, 256)
               const int* __restrict__ eidx,      // (2, E) flat: rows then cols
               float* __restrict__ out,           // (E,)
               int L)
{
    const int node = blockIdx.x * 16 + (threadIdx.x >> 4);
    if (node >= L) return;
    const int q16 = threadIdx.x & 15;
    const int j   = q16 & 7;
    const int h   = q16 >> 3;
    const size_t E = (size_t)L * DEG;
    const size_t e = (size_t)node * DEG + j;
    const int col  = eidx[E + e];                 // gather target node

    const float4* __restrict__ qp = (const float4*)(qk + (size_t)node * QK_COLS + h * 128);
    const float4* __restrict__ kp = (const float4*)(qk + (size_t)col  * QK_COLS + h * 128 + 64);

    float s = 0.0f;
    #pragma unroll
    for (int t = 0; t < 16; ++t) {
        float4 a = qp[t], b = kp[t];
        s = fmaf(a.x, b.x, s);
        s = fmaf(a.y, b.y, s);
        s = fmaf(a.z, b.z, s);
        s = fmaf(a.w, b.w, s);
    }
    s *= 8.0f;                                    // reference: s / (1/sqrt(64))

    // max over the 8 edges of this (node, head): xor 1,2,4 stays in 8-lane group
    float m = s;
    m = fmaxf(m, __shfl_xor(m, 1));
    m = fmaxf(m, __shfl_xor(m, 2));
    m = fmaxf(m, __shfl_xor(m, 4));
    float ev = __expf(s - m);
    float z = ev;
    z += __shfl_xor(z, 1);
    z += __shfl_xor(z, 2);
    z += __shfl_xor(z, 4);
    float attn = ev / z;

    // mean over heads: xor 8 swaps the two head halves of the 16-lane node group
    float mean = 0.5f * (attn + __shfl_xor(attn, 8));
    if (h == 0) out[e] = mean;
}

extern "C" void kernel_launch(void* const* d_in, const int* in_sizes, int n_in,
                              void* d_out, int out_size, void* d_ws, size_t ws_size,
                              hipStream_t stream) {
    const float* x    = (const float*)d_in[0];    // (L, 64)
    const float* W    = (const float*)d_in[1];    // (384, 64)
    const float* b    = (const float*)d_in[2];    // (384,)
    const int*   eidx = (const int*)d_in[3];      // (2, L*8)
    float* out = (float*)d_out;
    float* qk  = (float*)d_ws;                    // needs L*256*4 bytes (102.4 MB)

    const int L = in_sizes[0] / DM;
    const int mtiles = (L + 15) / 16;

    qk_gemm_wmma<<<mtiles, 512, 0, stream>>>(x, W, b, qk, L);
    edge_attn  <<<mtiles, 256, 0, stream>>>(qk, eidx, out, L);
}